// inputEncoder_62526133895566
// MI455X (gfx1250) — compile-verified
//
#include <hip/hip_runtime.h>
#include <hip/hip_bf16.h>
#include <math.h>

#define EPSV 1e-07f

typedef __attribute__((ext_vector_type(2))) float v2f;
typedef __attribute__((ext_vector_type(8))) float v8f;
typedef __attribute__((ext_vector_type(4))) int v4i;

// ---------------------------------------------------------------------------
// CDNA5 async global->LDS copy (ASYNCcnt path), with safe fallback.
// Probe result: __builtin_amdgcn_global_load_async_to_lds_b128 takes
// (int4 AS1*, int4 AS3*, imm offset, cpol).
// ---------------------------------------------------------------------------
#if __has_builtin(__builtin_amdgcn_global_load_async_to_lds_b128) && \
    __has_builtin(__builtin_amdgcn_s_wait_asynccnt)
#define HAVE_ASYNC_LDS 1
#else
#define HAVE_ASYNC_LDS 0
#endif

typedef __attribute__((address_space(1))) v4i as1_v4i;
typedef __attribute__((address_space(3))) v4i as3_v4i;

__device__ __forceinline__ void cp16_async(const float* g, float* l) {
#if HAVE_ASYNC_LDS
    __builtin_amdgcn_global_load_async_to_lds_b128(
        (as1_v4i*)(void*)g, (as3_v4i*)l, 0, 0);
#else
    *(float4*)l = *(const float4*)g;
#endif
}

__device__ __forceinline__ void wait_async_all() {
#if HAVE_ASYNC_LDS
    __builtin_amdgcn_s_wait_asynccnt(0);
#endif
}

// ---------------------------------------------------------------------------
// Kernel A: row-normalize encoder_vector (K=64) and emit output affine params.
// One wave32 per row, 2 elements per lane, shfl_xor tree reductions.
// ---------------------------------------------------------------------------
__global__ void normalize_rows(const float* __restrict__ x,
                               float* __restrict__ ev,
                               float* __restrict__ scale,
                               float* __restrict__ bias) {
    const int row  = blockIdx.x * 8 + (threadIdx.x >> 5);  // 8 waves per block
    const int lane = threadIdx.x & 31;
    const float* xr = x + (size_t)row * 64;

    float v0 = xr[lane];
    float v1 = xr[lane + 32];

    float s = v0 + v1;
    #pragma unroll
    for (int off = 16; off; off >>= 1) s += __shfl_xor(s, off, 32);
    const float mean = s * (1.0f / 64.0f);

    float d0 = v0 - mean, d1 = v1 - mean;
    float ss = d0 * d0 + d1 * d1;
    #pragma unroll
    for (int off = 16; off; off >>= 1) ss += __shfl_xor(ss, off, 32);
    const float stdv = sqrtf(ss * (1.0f / 63.0f));        // ddof=1
    const float inv  = 1.0f / (stdv + EPSV);

    const float e0 = d0 * inv, e1 = d1 * inv;
    ev[(size_t)row * 64 + lane]      = e0;
    ev[(size_t)row * 64 + lane + 32] = e1;

    // Faithful: stats of the *normalized* vector drive the output affine.
    float s2 = e0 + e1;
    #pragma unroll
    for (int off = 16; off; off >>= 1) s2 += __shfl_xor(s2, off, 32);
    const float meanT = s2 * (1.0f / 64.0f);
    float t0 = e0 - meanT, t1 = e1 - meanT;
    float ss2 = t0 * t0 + t1 * t1;
    #pragma unroll
    for (int off = 16; off; off >>= 1) ss2 += __shfl_xor(ss2, off, 32);
    const float stdT = sqrtf(ss2 * (1.0f / 63.0f));
    const float sc   = 1.0f / (stdT + EPSV);
    if (lane == 0) {
        scale[row] = sc;
        bias[row]  = -meanT * sc;
    }
}

// ---------------------------------------------------------------------------
// Kernel B: S[b] = ev[b] (1024x64) x ev[b]^T (64x1024), fp32 WMMA 16x16x4.
// 64x64 tile per 256-thread block; 8 waves, each produces 16x32 (2 accums).
// ---------------------------------------------------------------------------
__global__ void similarity_gemm(const float* __restrict__ ev,
                                float* __restrict__ S) {
    __shared__ float Ae[64][68];
    __shared__ float Be[64][68];

    const int b  = blockIdx.z;
    const int tm = blockIdx.y;
    const int tn = blockIdx.x;
    const float* evb = ev + (size_t)b * 1024 * 64;
    const int tid = threadIdx.x;

    for (int i = tid; i < 1024; i += 256) {            // 64*64/4 float4s
        const int row = i >> 4;
        const int c4  = (i & 15) << 2;
        cp16_async(evb + (size_t)(tm * 64 + row) * 64 + c4, &Ae[row][c4]);
        cp16_async(evb + (size_t)(tn * 64 + row) * 64 + c4, &Be[row][c4]);
    }
    wait_async_all();
    __syncthreads();

    const int lane = tid & 31;
    const int w    = tid >> 5;
    const int m0   = (w >> 1) * 16;
    const int n0   = (w & 1) * 32;
    const int r    = lane & 15;
    const int kh   = lane >> 4;

    v8f c0 = {}, c1 = {};
    #pragma unroll
    for (int k = 0; k < 64; k += 4) {
        v2f a  = *(const v2f*)&Ae[m0 + r][k + 2 * kh];
        v2f b0 = *(const v2f*)&Be[n0 + r][k + 2 * kh];
        v2f b1 = *(const v2f*)&Be[n0 + 16 + r][k + 2 * kh];
        c0 = __builtin_amdgcn_wmma_f32_16x16x4_f32(false, a, false, b0, (short)0, c0, false, false);
        c1 = __builtin_amdgcn_wmma_f32_16x16x4_f32(false, a, false, b1, (short)0, c1, false, false);
    }

    float* Sb = S + (size_t)b * 1024 * 1024;
    const int gmBase = tm * 64 + m0 + 8 * kh;
    const int gn     = tn * 64 + n0 + r;
    #pragma unroll
    for (int v = 0; v < 8; ++v) {
        Sb[(size_t)(gmBase + v) * 1024 + gn]      = c0[v];
        Sb[(size_t)(gmBase + v) * 1024 + gn + 16] = c1[v];
    }
}

// ---------------------------------------------------------------------------
// Kernel C: W[b] = S[b] (1024x1024) x T_flat[b] (1024x1024).
// 128x128 block tile, BK=16, double-buffered LDS with async global->LDS,
// 8 waves each computing 32x64 (8 fp32-WMMA accumulators).
// Fused per-row affine epilogue: out = W*scale[row] + bias[row].
// ---------------------------------------------------------------------------
__global__ void aggregate_gemm(const float* __restrict__ S,
                               const float* __restrict__ T,
                               const float* __restrict__ scale,
                               const float* __restrict__ bias,
                               float* __restrict__ out) {
    __shared__ float As[2][128][20];   // 128x16 tile/buffer, stride 80B
    __shared__ float Bs[2][16][136];   // 16x128 tile/buffer, stride 544B

    const int b  = blockIdx.z;
    const int tm = blockIdx.y;
    const int tn = blockIdx.x;
    const float* Sb = S + (size_t)b * 1024 * 1024;
    const float* Tb = T + (size_t)b * 1024 * 1024;
    const int tid  = threadIdx.x;
    const int lane = tid & 31;
    const int w    = tid >> 5;
    const int m0   = (w >> 1) * 32;    // 4 wave-rows of 32
    const int n0   = (w & 1) * 64;     // 2 wave-cols of 64
    const int r    = lane & 15;
    const int kh   = lane >> 4;

    // Per-thread staging coordinates.
    const int rA = tid >> 2;            // 0..63  (row of A tile; +64 for 2nd)
    const int cA = (tid & 3) << 2;      // 0..12  (col, float4)
    const int rB = tid >> 5;            // 0..7   (row of B tile; +8 for 2nd)
    const int cB = (tid & 31) << 2;     // 0..124 (col, float4)
    const float* gA = Sb + (size_t)(tm * 128 + rA) * 1024 + cA;
    const float* gB = Tb + (size_t)rB * 1024 + tn * 128 + cB;

    auto stage = [&](int k0, int buf) {
        cp16_async(gA + k0,                           &As[buf][rA][cA]);
        cp16_async(gA + k0 + (size_t)64 * 1024,       &As[buf][rA + 64][cA]);
        cp16_async(gB + (size_t)k0 * 1024,            &Bs[buf][rB][cB]);
        cp16_async(gB + (size_t)(k0 + 8) * 1024,      &Bs[buf][rB + 8][cB]);
    };

    v8f acc[2][4] = {};

    stage(0, 0);
    wait_async_all();
    __syncthreads();

    const int NT = 1024 / 16;
    for (int kt = 0; kt < NT; ++kt) {
        const int cur = kt & 1;
        if (kt + 1 < NT) stage((kt + 1) * 16, cur ^ 1);

        #pragma unroll
        for (int kk = 0; kk < 16; kk += 4) {
            const int krow = kk + 2 * kh;
            v2f a0 = *(const v2f*)&As[cur][m0 + r][krow];
            v2f a1 = *(const v2f*)&As[cur][m0 + 16 + r][krow];
            #pragma unroll
            for (int t = 0; t < 4; ++t) {
                v2f bf;
                bf.x = Bs[cur][krow][n0 + 16 * t + r];
                bf.y = Bs[cur][krow + 1][n0 + 16 * t + r];
                acc[0][t] = __builtin_amdgcn_wmma_f32_16x16x4_f32(
                    false, a0, false, bf, (short)0, acc[0][t], false, false);
                acc[1][t] = __builtin_amdgcn_wmma_f32_16x16x4_f32(
                    false, a1, false, bf, (short)0, acc[1][t], false, false);
            }
        }
        wait_async_all();
        __syncthreads();
    }

    // Fused epilogue: out = acc * scale[row] + bias[row]
    const float* scb = scale + b * 1024;
    const float* bib = bias + b * 1024;
    float* outb = out + (size_t)b * 1024 * 1024;
    #pragma unroll
    for (int mi = 0; mi < 2; ++mi) {
        const int gm0 = tm * 128 + m0 + mi * 16 + 8 * kh;
        #pragma unroll
        for (int v = 0; v < 8; ++v) {
            const int gm = gm0 + v;
            const float sc = scb[gm];
            const float bi = bib[gm];
            float* orow = outb + (size_t)gm * 1024 + tn * 128 + n0 + r;
            #pragma unroll
            for (int t = 0; t < 4; ++t)
                orow[16 * t] = acc[mi][t][v] * sc + bi;
        }
    }
}

// ---------------------------------------------------------------------------
extern "C" void kernel_launch(void* const* d_in, const int* in_sizes, int n_in,
                              void* d_out, int out_size, void* d_ws, size_t ws_size,
                              hipStream_t stream) {
    (void)in_sizes; (void)n_in; (void)out_size; (void)ws_size;
    const float* encoder_vector = (const float*)d_in[0];   // [8,1024,64]
    const float* encoder_T      = (const float*)d_in[1];   // [8,1024,32,32]
    float* out = (float*)d_out;                            // [8,1024,1024]

    // Workspace layout (floats): ev | scale | bias | S
    float* ev    = (float*)d_ws;                 // 8*1024*64  = 524288
    float* scale = ev + 8 * 1024 * 64;           // 8192
    float* bias  = scale + 8192;                 // 8192
    float* S     = bias + 8192;                  // 8*1024*1024 (16B aligned)

    normalize_rows<<<dim3(1024), dim3(256), 0, stream>>>(encoder_vector, ev, scale, bias);
    similarity_gemm<<<dim3(16, 16, 8), dim3(256), 0, stream>>>(ev, S);
    aggregate_gemm<<<dim3(8, 8, 8), dim3(256), 0, stream>>>(S, encoder_T, scale, bias, out);
}